// GCNProbe_22136261443777
// MI455X (gfx1250) — compile-verified
//
#include <hip/hip_runtime.h>
#include <hip/hip_bf16.h>
#include <math.h>

#define HDIM 128

typedef __attribute__((ext_vector_type(2))) float v2f;
typedef __attribute__((ext_vector_type(8))) float v8f;
typedef __attribute__((ext_vector_type(4))) unsigned int u32x4;
typedef __attribute__((ext_vector_type(8))) int i32x8;
typedef __attribute__((ext_vector_type(4))) int i32x4;

__device__ __forceinline__ float san0(float v) {
  // nan/inf -> 0 (values in this pipeline are bounded; matches ref intent)
  return isfinite(v) ? v : 0.0f;
}

// ---------------------------------------------------------------- zero fill
__global__ void k_zero(float* __restrict__ p, long n) {
  long i = (long)blockIdx.x * blockDim.x + threadIdx.x;
  if (i < n) p[i] = 0.0f;
}

// ------------------------------------------- edge weights sanitize + degree
__global__ void k_edge_prep(const float* __restrict__ w_in,
                            const int* __restrict__ dst,
                            float* __restrict__ ew,
                            float* __restrict__ deg, int E) {
  int e = blockIdx.x * blockDim.x + threadIdx.x;
  if (e < E) {
    float w = w_in[e];
    ew[e] = isfinite(w) ? fabsf(w) : 0.0f;
    atomicAdd(&deg[dst[e]], 1.0f);
  }
}

// ------------------------------------------------------- embedding gather
__global__ void k_gather_emb(const int* __restrict__ x,
                             const float* __restrict__ emb,
                             float* __restrict__ h, int N) {
  long i = (long)blockIdx.x * blockDim.x + threadIdx.x;
  int node = (int)(i >> 5);
  int c = ((int)i & 31) << 2;
  if (node < N) {
    float4 v = *(const float4*)(emb + (size_t)x[node] * HDIM + c);
    *(float4*)(h + (size_t)node * HDIM + c) = v;
  }
}

// ------------------------------------------------ edge scatter (L2 atomics)
// grid-stride loop, one wave32 per edge per iteration; each lane owns 4 ch.
// h and agg are L2-resident (51 MB each << 192 MB L2); prefetch next src row.
__global__ __launch_bounds__(256) void k_scatter(const float* __restrict__ h,
                                                 const float* __restrict__ ew,
                                                 const int* __restrict__ dst,
                                                 const int* __restrict__ src,
                                                 float* __restrict__ agg, int E) {
  const int lane = threadIdx.x & 31;
  const int stride = gridDim.x * 8;
  for (int e = blockIdx.x * 8 + (threadIdx.x >> 5); e < E; e += stride) {
    // prefetch the next iteration's gather row into the cache hierarchy
    int enext = e + stride;
    if (enext < E)
      __builtin_prefetch(h + (size_t)src[enext] * HDIM + lane * 4, 0, 0);
    float w = ew[e];
    int s = src[e], d = dst[e];
    const float4 v = *(const float4*)(h + (size_t)s * HDIM + lane * 4);
    float* o = agg + (size_t)d * HDIM + lane * 4;
    atomicAdd(o + 0, san0(v.x) * w);
    atomicAdd(o + 1, san0(v.y) * w);
    atomicAdd(o + 2, san0(v.z) * w);
    atomicAdd(o + 3, san0(v.w) * w);
  }
}

// --------------------------------- fused (agg/deg) @ W + bias, ReLU via WMMA
// block: 256 threads = 8 waves; each block computes 128 rows x 128 cols.
// W (128x128 f32, 64KB) staged into LDS by the Tensor Data Mover (one DMA
// descriptor, wave 0 issues, s_wait_tensorcnt + barrier). A operand rows are
// preloaded into registers from L2-resident agg, then 256 f32 WMMAs run with
// only LDS traffic in the inner loop.
__global__ __launch_bounds__(256) void k_gemm_relu_wmma(
    const float* __restrict__ agg, const float* __restrict__ deg,
    const float* __restrict__ W, const float* __restrict__ bias,
    float* __restrict__ hout, int N) {
  __shared__ float Ws[HDIM * HDIM];  // exactly 64KB, LDS offset 0

  const int tid = threadIdx.x;

#if defined(__has_builtin) && __has_builtin(__builtin_amdgcn_tensor_load_to_lds)
  // ---- TDM stage of W: D# per CDNA5 ISA ch.8 (2D tensor, tile == tensor)
  if (tid < 32) {  // only wave 0 issues the DMA (TDM ignores EXEC)
    unsigned long long ga = (unsigned long long)W;
    u32x4 g0;
    g0.x = 1u;                                   // count=1, is_restore=0
    g0.y = 0u;                                   // lds_addr = 0 (Ws base)
    g0.z = (unsigned int)(ga & 0xffffffffull);   // global_addr[31:0]
    g0.w = (unsigned int)((ga >> 32) & 0x01ffffffull)  // global_addr[56:32]
           | (2u << 30);                         // type = 2 ("image")
    i32x8 g1;
    g1[0] = (int)(2u << 16);    // workgroup_mask=0, data_size=2 (4 bytes)
    g1[1] = (int)(128u << 16);  // tensor_dim0 = 128
    g1[2] = (int)(128u << 16);  // tensor_dim1 = 128
    g1[3] = (int)(128u << 16);  // tile_dim0 = 128
    g1[4] = 128;                // tile_dim1 = 128
    g1[5] = 128;                // tensor_dim0_stride = 128
    g1[6] = 0;
    g1[7] = 0;
    i32x4 zg4 = {0, 0, 0, 0};   // groups 2/3: higher dims unused (tile_dim=0)
    i32x8 zg8 = {0, 0, 0, 0, 0, 0, 0, 0};
    __builtin_amdgcn_tensor_load_to_lds(g0, g1, zg4, zg4, zg8, 0);
    __builtin_amdgcn_s_wait_tensorcnt(0);
  }
  // The compiler cannot see that the TDM wrote Ws (the builtin only takes a
  // descriptor). Escape the LDS pointer through an opaque asm with a memory
  // clobber so the B-fragment ds_loads below are not folded to undef/DCE'd.
  {
    void* pw = (void*)Ws;
    asm volatile("" : "+v"(pw) : : "memory");
  }
#else
  for (int i = tid; i < HDIM * HDIM / 4; i += 256)
    ((float4*)Ws)[i] = ((const float4*)W)[i];
#endif
  __syncthreads();

  const int wave  = tid >> 5;
  const int lane  = tid & 31;
  const int m     = lane & 15;          // M (A frag) / N (B,D frags)
  const int khalf = (lane >> 4) << 1;   // 0 for lanes 0-15, 2 for lanes 16-31

  const int  row0  = blockIdx.x * 128;
  const int  grow  = row0 + wave * 16 + m;
  const bool rowok = grow < N;
  const float* Arow = agg + (size_t)(rowok ? grow : 0) * HDIM;
  const float  sc   = rowok ? (1.0f / fmaxf(deg[grow], 1.0f)) : 0.0f;

  // preload this lane's K-half of its A row (pairs every 4 columns),
  // sanitize + degree-normalize once; 32 b64 loads issued back-to-back
  v2f areg[HDIM / 4];
#pragma unroll
  for (int t = 0; t < HDIM / 4; ++t) {
    v2f a;
    a.x = Arow[4 * t + khalf];
    a.y = Arow[4 * t + khalf + 1];
    areg[t] = a;
  }
#pragma unroll
  for (int t = 0; t < HDIM / 4; ++t) {
    areg[t].x = san0(areg[t].x) * sc;
    areg[t].y = san0(areg[t].y) * sc;
  }

  v8f acc[8] = {};  // 8 column tiles of 16x16

#pragma unroll
  for (int t = 0; t < HDIM / 4; ++t) {
    const int k0 = 4 * t;
    const v2f a = areg[t];
    const float* Brow = &Ws[(k0 + khalf) * HDIM + m];
#pragma unroll
    for (int nt = 0; nt < 8; ++nt) {
      // B frag 4x16 f32: rows (k0+khalf, k0+khalf+1), cols nt*16+m
      v2f b;
      b.x = Brow[nt * 16];
      b.y = Brow[nt * 16 + HDIM];
      acc[nt] = __builtin_amdgcn_wmma_f32_16x16x4_f32(
          false, a, false, b, (short)0, acc[nt], false, false);
    }
  }

  // D layout: VGPR r -> M = r + (lane>=16 ? 8 : 0), N = (lane&15) + nt*16
  const int rbase = row0 + wave * 16 + ((lane >> 4) << 3);
#pragma unroll
  for (int nt = 0; nt < 8; ++nt) {
    const int col = nt * 16 + m;
    const float bval = bias[col];
#pragma unroll
    for (int r = 0; r < 8; ++r) {
      const int orow = rbase + r;
      if (orow < N) {
        float v = fmaxf(acc[nt][r] + bval, 0.0f);
        hout[(size_t)orow * HDIM + col] = san0(v);
      }
    }
  }
}

// ------------------------------------------------- per-graph mean/max pooling
// one wave32 per node; h >= 0 post-ReLU so integer atomicMax on bits is exact
__global__ __launch_bounds__(256) void k_pool(const float* __restrict__ h,
                                              const int* __restrict__ batch,
                                              float* __restrict__ gsum,
                                              float* __restrict__ gmax,
                                              float* __restrict__ gcnt, int N) {
  int i = blockIdx.x * 8 + (threadIdx.x >> 5);
  int lane = threadIdx.x & 31;
  if (i < N) {
    int g = batch[i];
    const float4 v = *(const float4*)(h + (size_t)i * HDIM + lane * 4);
    float* s = gsum + (size_t)g * HDIM + lane * 4;
    atomicAdd(s + 0, v.x);
    atomicAdd(s + 1, v.y);
    atomicAdd(s + 2, v.z);
    atomicAdd(s + 3, v.w);
    int* mx = (int*)(gmax + (size_t)g * HDIM + lane * 4);
    atomicMax(mx + 0, __float_as_int(v.x));
    atomicMax(mx + 1, __float_as_int(v.y));
    atomicMax(mx + 2, __float_as_int(v.z));
    atomicMax(mx + 3, __float_as_int(v.w));
    if (lane == 0) atomicAdd(&gcnt[g], 1.0f);
  }
}

// --------------------------------------------------------- tiny MLP head
__global__ __launch_bounds__(128) void k_head(const float* __restrict__ gsum,
                                              const float* __restrict__ gmax,
                                              const float* __restrict__ gcnt,
                                              const float* __restrict__ fc1w,
                                              const float* __restrict__ fc1b,
                                              const float* __restrict__ fc2w,
                                              const float* __restrict__ fc2b,
                                              float* __restrict__ out) {
  __shared__ float zs[HDIM];
  int g = blockIdx.x, j = threadIdx.x;
  float cnt = fmaxf(gcnt[g], 1.0f);
  const float* ms = gsum + (size_t)g * HDIM;
  const float* mx = gmax + (size_t)g * HDIM;
  float acc = fc1b[j];
  for (int k = 0; k < HDIM; ++k) acc += (ms[k] / cnt) * fc1w[k * HDIM + j];
  for (int k = 0; k < HDIM; ++k) acc += mx[k] * fc1w[(HDIM + k) * HDIM + j];
  zs[j] = fmaxf(acc, 0.0f) * fc2w[j];
  __syncthreads();
  for (int s = 64; s > 0; s >>= 1) {
    if (j < s) zs[j] += zs[j + s];
    __syncthreads();
  }
  if (j == 0) out[g] = zs[0] + fc2b[0];
}

// ---------------------------------------------------------------- launcher
extern "C" void kernel_launch(void* const* d_in, const int* in_sizes, int n_in,
                              void* d_out, int out_size, void* d_ws, size_t ws_size,
                              hipStream_t stream) {
  const int*   x     = (const int*)d_in[0];
  const int*   ei    = (const int*)d_in[1];
  const float* ewin  = (const float*)d_in[2];
  const int*   batch = (const int*)d_in[3];
  const float* emb   = (const float*)d_in[4];
  const float* Wc    = (const float*)d_in[5];
  const float* bc    = (const float*)d_in[6];
  const float* fc1w  = (const float*)d_in[7];
  const float* fc1b  = (const float*)d_in[8];
  const float* fc2w  = (const float*)d_in[9];
  const float* fc2b  = (const float*)d_in[10];
  float* out = (float*)d_out;

  const int N = in_sizes[0];
  const int E = in_sizes[1] / 2;
  const int G = out_size;
  const int L = in_sizes[5] / (HDIM * HDIM);
  const int* dst = ei;       // edge_index[0]
  const int* src = ei + E;   // edge_index[1]

  // workspace layout (floats)
  float* p    = (float*)d_ws;
  float* deg  = p; p += N;
  float* ew   = p; p += E;
  float* h    = p; p += (size_t)N * HDIM;
  float* agg  = p; p += (size_t)N * HDIM;
  float* gsum = p; p += (size_t)G * HDIM;
  float* gmax = p; p += (size_t)G * HDIM;   // contiguous with gsum
  float* gcnt = p; p += G;                  // contiguous with gmax
  (void)ws_size; (void)n_in;

  auto nblk = [](long n, long per) { return (int)((n + per - 1) / per); };

  // init accumulators
  k_zero<<<nblk(N, 256), 256, 0, stream>>>(deg, N);
  k_zero<<<nblk((long)G * HDIM * 2 + G, 256), 256, 0, stream>>>(
      gsum, (long)G * HDIM * 2 + G);

  k_edge_prep<<<nblk(E, 256), 256, 0, stream>>>(ewin, dst, ew, deg, E);
  k_gather_emb<<<nblk((long)N * 32, 256), 256, 0, stream>>>(x, emb, h, N);

  int sc_blocks = nblk(E, 8);
  if (sc_blocks > 8192) sc_blocks = 8192;  // grid-stride with prefetch

  for (int l = 0; l < L; ++l) {
    k_zero<<<nblk((long)N * HDIM, 256), 256, 0, stream>>>(agg, (long)N * HDIM);
    k_scatter<<<sc_blocks, 256, 0, stream>>>(h, ew, dst, src, agg, E);
    k_gemm_relu_wmma<<<nblk(N, 128), 256, 0, stream>>>(
        agg, deg, Wc + (size_t)l * HDIM * HDIM, bc + (size_t)l * HDIM, h, N);
  }

  k_pool<<<nblk(N, 8), 256, 0, stream>>>(h, batch, gsum, gmax, gcnt, N);
  k_head<<<G, 128, 0, stream>>>(gsum, gmax, gcnt, fc1w, fc1b, fc2w, fc2b, out);
}